// LeeOscillator_1915555414850
// MI455X (gfx1250) — compile-verified
//
#include <hip/hip_runtime.h>

#define N_STEPS 100
#define E_PERT 0.001f
// -K_GATE * log2(e) = -500 * 1.4426950408889634
#define NEG_K_LOG2E -721.3475204444817f

typedef float v2f __attribute__((ext_vector_type(2)));

// (a1,a2,a3,a4, b1,b2,b3,b4, xi_E, xi_I) per oscillator type; row 0 unused.
__device__ __constant__ float kCoef[9][10] = {
    { 0.0f,  0.0f,  0.0f,  0.0f,  0.0f,  0.0f,  0.0f,  0.0f, 0.0f, 0.0f},
    { 0.0f,  5.0f,  5.0f,  1.0f,  0.0f, -1.0f,  1.0f,  0.0f, 0.0f, 0.0f},
    { 0.5f,  0.55f, 0.55f,-0.5f,  0.5f, -0.55f,-0.55f,-0.5f, 0.0f, 0.0f},
    {-5.0f,  5.0f,  5.0f, -5.0f,  1.0f, -1.0f, -1.0f,  1.0f, 0.0f, 0.0f},
    { 1.0f,  1.0f,  1.0f, -1.0f, -1.0f, -1.0f, -1.0f,  1.0f, 0.0f, 0.0f},
    { 5.0f, -5.0f, -5.0f,  5.0f, -1.0f,  1.0f,  1.0f, -1.0f, 0.0f, 0.0f},
    {-1.0f, -1.0f, -1.0f,  1.0f,  1.0f,  1.0f,  1.0f, -1.0f, 0.0f, 0.0f},
    { 1.0f, -1.0f, -1.0f,  1.0f, -1.0f,  1.0f,  1.0f, -1.0f, 0.0f, 0.0f},
    {-1.0f,  1.0f,  1.0f, -1.0f,  1.0f, -1.0f, -1.0f,  1.0f, 0.0f, 0.0f},
};

// gfx1250 hardware tanh (CDNA5 trans op, co-executes with VALU).
__device__ __forceinline__ float tanh_hw(float x) {
#if defined(__HIP_DEVICE_COMPILE__)
#if __has_builtin(__builtin_amdgcn_tanhf)
  return __builtin_amdgcn_tanhf(x);
#elif __has_builtin(__builtin_amdgcn_tanh_f32)
  return __builtin_amdgcn_tanh_f32(x);
#else
  float r;
  asm("v_tanh_f32 %0, %1\n\tv_nop" : "=v"(r) : "v"(x));
  return r;
#endif
#else
  return tanhf(x);  // host semantic-check only; never codegen'd
#endif
}

__device__ __forceinline__ float exp2_hw(float x) {
#if defined(__HIP_DEVICE_COMPILE__) && __has_builtin(__builtin_amdgcn_exp2f)
  return __builtin_amdgcn_exp2f(x);
#else
  return exp2f(x);  // HIP __device__ overload on device; host checked only
#endif
}

__device__ __forceinline__ v2f vfma2(v2f a, v2f b, v2f c) {
#if __has_builtin(__builtin_elementwise_fma)
  return __builtin_elementwise_fma(a, b, c);
#else
  v2f r;
  r.x = fmaf(a.x, b.x, c.x);
  r.y = fmaf(a.y, b.y, c.y);
  return r;
#endif
}

__device__ __forceinline__ float lee_setup_sim(float xv) {
  float sg = (xv > 0.0f) ? 1.0f : ((xv < 0.0f) ? -1.0f : 0.0f);
  return fmaf(sg, E_PERT, xv);
}

// M2 packed (float2) chains = 2*M2 independent 99-step recurrences, interleaved
// so hardware-tanh latency is hidden and FMAs can select V_PK_FMA_F32 / VOPD.
template <int M2>
__device__ __forceinline__ void lee_run_pk(const float* __restrict__ xin,
                                           float* __restrict__ lout,
                                           float A1, float A2, float nA3, float A4,
                                           float B1, float nB2, float nB3, float B4,
                                           float XE, float XI) {
  const v2f vA1  = {A1, A1},   vA2  = {A2, A2},   vnA3 = {nA3, nA3};
  const v2f vB1  = {B1, B1},   vnB2 = {nB2, nB2}, vnB3 = {nB3, nB3};

  v2f simA[M2], simB[M2], gate[M2], omega[M2], E[M2], I[M2], L[M2];
#pragma unroll
  for (int k = 0; k < M2; ++k) {
    float s0 = lee_setup_sim(xin[2 * k + 0]);
    float s1 = lee_setup_sim(xin[2 * k + 1]);
    gate[k].x  = exp2_hw(NEG_K_LOG2E * s0 * s0);  // exp(-500*sim^2), step-invariant
    gate[k].y  = exp2_hw(NEG_K_LOG2E * s1 * s1);
    omega[k].x = tanh_hw(5.0f * s0);              // tanh(5*sim),    step-invariant
    omega[k].y = tanh_hw(5.0f * s1);
    simA[k].x  = fmaf(A4, s0, -XE);               // fold a4*sim - 5*xi_E
    simA[k].y  = fmaf(A4, s1, -XE);
    simB[k].x  = fmaf(B4, s0, -XI);               // fold b4*sim - 5*xi_I
    simB[k].y  = fmaf(B4, s1, -XI);
    E[k] = (v2f){0.2f, 0.2f};
    I[k] = (v2f){0.0f, 0.0f};
    L[k] = (v2f){0.2f, 0.2f};
  }

  for (int s = 0; s < N_STEPS - 1; ++s) {
#pragma unroll
    for (int k = 0; k < M2; ++k) {
      v2f eA = vfma2(vA1,  L[k], simA[k]);
      eA     = vfma2(vA2,  E[k], eA);
      eA     = vfma2(vnA3, I[k], eA);
      v2f iA = vfma2(vB1,  L[k], simB[k]);
      iA     = vfma2(vnB2, E[k], iA);
      iA     = vfma2(vnB3, I[k], iA);
      v2f E1, I1;
      E1.x = tanh_hw(eA.x);
      E1.y = tanh_hw(eA.y);
      I1.x = tanh_hw(iA.x);
      I1.y = tanh_hw(iA.y);
      L[k] = vfma2(E1 - I1, gate[k], omega[k]);
      E[k] = E1;
      I[k] = I1;
    }
  }

#pragma unroll
  for (int k = 0; k < M2; ++k) {
    lout[2 * k + 0] = L[k].x;
    lout[2 * k + 1] = L[k].y;
  }
}

// Scalar version for the guarded tail path.
__device__ __forceinline__ float lee_run_1(float xv,
                                           float A1, float A2, float nA3, float A4,
                                           float B1, float nB2, float nB3, float B4,
                                           float XE, float XI) {
  float sim   = lee_setup_sim(xv);
  float gate  = exp2_hw(NEG_K_LOG2E * sim * sim);
  float omega = tanh_hw(5.0f * sim);
  float simA  = fmaf(A4, sim, -XE);
  float simB  = fmaf(B4, sim, -XI);
  float E = 0.2f, I = 0.0f, L = 0.2f;
  for (int s = 0; s < N_STEPS - 1; ++s) {
    float eA = fmaf(A1,  L, simA);
    eA       = fmaf(A2,  E, eA);
    eA       = fmaf(nA3, I, eA);
    float iA = fmaf(B1,  L, simB);
    iA       = fmaf(nB2, E, iA);
    iA       = fmaf(nB3, I, iA);
    float E1 = tanh_hw(eA);
    float I1 = tanh_hw(iA);
    L = fmaf(E1 - I1, gate, omega);
    E = E1;
    I = I1;
  }
  return L;
}

__global__ __launch_bounds__(256) void lee_osc_kernel(
    const float* __restrict__ x, const int* __restrict__ otype_p,
    float* __restrict__ out, int n) {
  // Runtime oscillator type (device scalar) -> uniform coefficient fetch.
  int t = otype_p[0];
  if (t < 1 || t > 8) t = 1;
  // Fold tanh5's *5 into all coefficients; pre-negate a3,b2,b3 for pure FMAs.
  const float A1 = 5.0f * kCoef[t][0];
  const float A2 = 5.0f * kCoef[t][1];
  const float nA3 = -5.0f * kCoef[t][2];
  const float A4 = 5.0f * kCoef[t][3];
  const float B1 = 5.0f * kCoef[t][4];
  const float nB2 = -5.0f * kCoef[t][5];
  const float nB3 = -5.0f * kCoef[t][6];
  const float B4 = 5.0f * kCoef[t][7];
  const float XE = 5.0f * kCoef[t][8];
  const float XI = 5.0f * kCoef[t][9];

  constexpr int EPT = 8;  // elements per thread (4 packed float2 chains)
  const long long base = (long long)(blockIdx.x * blockDim.x + threadIdx.x) * EPT;

  if (base + EPT <= n) {
    // Fast path: two 128-bit loads, 4 packed chains, two 128-bit stores.
    float buf[EPT];
    const float4* __restrict__ xin4 = reinterpret_cast<const float4*>(x + base);
    float4 v0 = xin4[0];
    float4 v1 = xin4[1];
    buf[0] = v0.x; buf[1] = v0.y; buf[2] = v0.z; buf[3] = v0.w;
    buf[4] = v1.x; buf[5] = v1.y; buf[6] = v1.z; buf[7] = v1.w;

    float lf[EPT];
    lee_run_pk<EPT / 2>(buf, lf, A1, A2, nA3, A4, B1, nB2, nB3, B4, XE, XI);

    float4* __restrict__ out4 = reinterpret_cast<float4*>(out + base);
    out4[0] = make_float4(lf[0], lf[1], lf[2], lf[3]);
    out4[1] = make_float4(lf[4], lf[5], lf[6], lf[7]);
  } else {
    // Tail path (not taken for 4096x4096): scalar, guarded.
    for (int k = 0; k < EPT; ++k) {
      long long i = base + k;
      if (i < n) {
        out[i] = lee_run_1(x[i], A1, A2, nA3, A4, B1, nB2, nB3, B4, XE, XI);
      }
    }
  }
}

extern "C" void kernel_launch(void* const* d_in, const int* in_sizes, int n_in,
                              void* d_out, int out_size, void* d_ws, size_t ws_size,
                              hipStream_t stream) {
  (void)n_in; (void)out_size; (void)d_ws; (void)ws_size;
  const float* x = (const float*)d_in[0];
  const int* otype = (const int*)d_in[1];
  float* out = (float*)d_out;
  const int n = in_sizes[0];

  constexpr int THREADS = 256;
  constexpr int EPT = 8;
  const int per_block = THREADS * EPT;
  const int blocks = (n + per_block - 1) / per_block;

  lee_osc_kernel<<<blocks, THREADS, 0, stream>>>(x, otype, out, n);
}